// DynamicLocal_9423158248007
// MI455X (gfx1250) — compile-verified
//
#include <hip/hip_runtime.h>

#define B_  8
#define C_  128
#define H_  128
#define W_  128
#define HW_ (H_ * W_)
#define KK  9
#define KO  (C_ * KK)   // 1152

typedef float v2f __attribute__((ext_vector_type(2)));
typedef float v4f __attribute__((ext_vector_type(4)));
typedef float v8f __attribute__((ext_vector_type(8)));

// ---------------------------------------------------------------------------
// Kernel 1: global average pool  x[b,c,:,:] -> pooled[b*C+c]
// Also zero-fills the padding rows 8..15 of the padded [16 x 128] pooled mat,
// and prefetches w1/w2 into L2 (global_prefetch_b8) so the single-workgroup
// generator GEMM that follows is L2-latency-bound, not HBM-latency-bound.
// ---------------------------------------------------------------------------
__global__ void __launch_bounds__(256)
pool_kernel(const float* __restrict__ x, float* __restrict__ pooled,
            const float* __restrict__ w1, const float* __restrict__ w2)
{
    const int plane = blockIdx.x;                 // b*C + c  (0..1023)
    const int t     = threadIdx.x;
    const float4* xp = (const float4*)(x + (size_t)plane * HW_);

    // Warm L2 with the generator weights: w1 = 512 lines, w2 = 4608 lines.
    const int g = plane * 256 + t;                // global thread id
    if (g < 512)        __builtin_prefetch((const char*)w1 + (size_t)g * 128);
    else if (g < 5120)  __builtin_prefetch((const char*)w2 + (size_t)(g - 512) * 128);

    float s = 0.f;
#pragma unroll
    for (int i = 0; i < 16; ++i) {
        float4 v = xp[t + i * 256];
        s += (v.x + v.y) + (v.z + v.w);
    }

    __shared__ float red[256];
    red[t] = s;
    __syncthreads();
    for (int off = 128; off > 0; off >>= 1) {
        if (t < off) red[t] += red[t + off];
        __syncthreads();
    }
    if (t == 0) pooled[plane] = red[0] * (1.0f / (float)HW_);
    if (t == 1) pooled[B_ * C_ + plane] = 0.f;    // zero pad rows 8..15
}

// ---------------------------------------------------------------------------
// Kernel 2: filter generation with fp32 WMMA (V_WMMA_F32_16X16X4_F32).
//   h    = relu(Ppad[16x128] @ w1^T + b1)       (Ppad rows 8..15 are zero)
//   kern = h @ w2^T + b2  -> [16 x 1152]        (rows 0..7 consumed)
// One workgroup, 8 waves; each wave owns 16-wide N tiles; K-loop = 32 chained
// accumulating WMMAs per tile. A/B/C fragment layouts per CDNA5 ISA 7.12.2.
// ---------------------------------------------------------------------------
__global__ void __launch_bounds__(256)
kgen_kernel(const float* __restrict__ pooled,
            const float* __restrict__ w1, const float* __restrict__ b1,
            const float* __restrict__ w2, const float* __restrict__ b2,
            float* __restrict__ kern)
{
    __shared__ float hsh[16 * C_];

    const int t    = threadIdx.x;
    const int wave = t >> 5;
    const int lane = t & 31;
    const int half = lane >> 4;       // 0: lanes 0-15, 1: lanes 16-31
    const int mn   = lane & 15;       // A row (M) / B,C,D column (N)

    // ---- stage A: h = relu(Ppad @ w1^T + b1), one 16x16 tile per wave ----
    {
        const int o0 = wave * 16;
        v8f c = {};
        for (int k = 0; k < C_; k += 4) {
            // A[m][k + 2*half + v]
            v2f a = *(const v2f*)(pooled + mn * C_ + k + 2 * half);
            // B[kk][n] = w1[o0+n][k+kk]  ->  b[v] = w1[o0+n][k + 2*half + v]
            v2f b = *(const v2f*)(w1 + (size_t)(o0 + mn) * C_ + k + 2 * half);
            c = __builtin_amdgcn_wmma_f32_16x16x4_f32(
                    false, a, false, b, (short)0, c, false, false);
        }
        const float bias = b1[o0 + mn];
#pragma unroll
        for (int r = 0; r < 8; ++r) {             // D[r + 8*half][n]
            float v = c[r] + bias;
            hsh[(r + 8 * half) * C_ + o0 + mn] = v > 0.f ? v : 0.f;
        }
    }
    __syncthreads();

    // ---- stage B: kern = h @ w2^T + b2; 72 tiles, wave w -> tiles w+8*i ----
    for (int i = 0; i < 9; ++i) {
        const int o0 = (wave + 8 * i) * 16;
        v8f c = {};
        for (int k = 0; k < C_; k += 4) {
            v2f a = *(const v2f*)(&hsh[mn * C_ + k + 2 * half]);
            v2f b = *(const v2f*)(w2 + (size_t)(o0 + mn) * C_ + k + 2 * half);
            c = __builtin_amdgcn_wmma_f32_16x16x4_f32(
                    false, a, false, b, (short)0, c, false, false);
        }
        const float bias = b2[o0 + mn];
#pragma unroll
        for (int r = 0; r < 8; ++r)
            kern[(size_t)(r + 8 * half) * KO + o0 + mn] = c[r] + bias;
    }
}

// ---------------------------------------------------------------------------
// Kernel 3: dynamic depthwise 3x3 conv (cross-correlation, zero pad).
// One block per (b,c) plane. Thread owns a 4-wide x 16-tall strip and walks
// down with a register sliding window (each row fetched once: b128 + 2*b32).
// Output stores are non-temporal so the 64 MiB out-stream does not evict x
// from L2 (x is read again here after the pooling pass).
// ---------------------------------------------------------------------------
__device__ __forceinline__ void load_row6(const float* __restrict__ xp,
                                          int y, int cq, float w[6])
{
    if ((unsigned)y < (unsigned)H_) {
        const float* r = xp + y * W_ + cq;
        float4 m = *(const float4*)r;             // 16B aligned (cq % 4 == 0)
        w[1] = m.x; w[2] = m.y; w[3] = m.z; w[4] = m.w;
        w[0] = (cq > 0)      ? r[-1] : 0.f;
        w[5] = (cq < W_ - 4) ? r[4]  : 0.f;
    } else {
#pragma unroll
        for (int j = 0; j < 6; ++j) w[j] = 0.f;
    }
}

__global__ void __launch_bounds__(256)
dwconv_kernel(const float* __restrict__ x, const float* __restrict__ kern,
              float* __restrict__ out)
{
    const int plane = blockIdx.x;                 // b*C + c
    const int b = plane >> 7;
    const int c = plane & (C_ - 1);

    const float* kp = kern + (size_t)b * KO + c * KK;   // block-uniform
    const float k0 = kp[0], k1 = kp[1], k2 = kp[2];
    const float k3 = kp[3], k4 = kp[4], k5 = kp[5];
    const float k6 = kp[6], k7 = kp[7], k8 = kp[8];

    const float* xp = x   + (size_t)plane * HW_;
    float*       op = out + (size_t)plane * HW_;

    const int t  = threadIdx.x;
    const int cq = (t & 31) * 4;                  // column strip start
    const int r0 = (t >> 5) * 16;                 // row strip start

    float wm1[6], w0[6], wp1[6];
    load_row6(xp, r0 - 1, cq, wm1);
    load_row6(xp, r0,     cq, w0);

    for (int y = r0; y < r0 + 16; ++y) {
        load_row6(xp, y + 1, cq, wp1);
        v4f o;
#pragma unroll
        for (int j = 0; j < 4; ++j) {
            float acc;
            acc = k0 * wm1[j];
            acc = fmaf(k1, wm1[j + 1], acc);
            acc = fmaf(k2, wm1[j + 2], acc);
            acc = fmaf(k3, w0 [j    ], acc);
            acc = fmaf(k4, w0 [j + 1], acc);
            acc = fmaf(k5, w0 [j + 2], acc);
            acc = fmaf(k6, wp1[j    ], acc);
            acc = fmaf(k7, wp1[j + 1], acc);
            acc = fmaf(k8, wp1[j + 2], acc);
            o[j] = acc;
        }
        __builtin_nontemporal_store(o, (v4f*)(op + y * W_ + cq));
#pragma unroll
        for (int j = 0; j < 6; ++j) { wm1[j] = w0[j]; w0[j] = wp1[j]; }
    }
}

// ---------------------------------------------------------------------------
extern "C" void kernel_launch(void* const* d_in, const int* in_sizes, int n_in,
                              void* d_out, int out_size, void* d_ws, size_t ws_size,
                              hipStream_t stream)
{
    const float* x  = (const float*)d_in[0];
    const float* w1 = (const float*)d_in[1];
    const float* b1 = (const float*)d_in[2];
    const float* w2 = (const float*)d_in[3];
    const float* b2 = (const float*)d_in[4];
    float* out = (float*)d_out;

    float* pooled = (float*)d_ws;                 // [16 x 128] (rows 8..15 = 0)
    float* kern   = pooled + 16 * C_;             // [16 x 1152]

    pool_kernel  <<<B_ * C_, 256, 0, stream>>>(x, pooled, w1, w2);
    kgen_kernel  <<<1,       256, 0, stream>>>(pooled, w1, b1, w2, b2, kern);
    dwconv_kernel<<<B_ * C_, 256, 0, stream>>>(x, kern, out);
}